// AttnDecoderRNN_63359357551159
// MI455X (gfx1250) — compile-verified
//
#include <hip/hip_runtime.h>
#include <hip/hip_bf16.h>

// Sizes from the reference
constexpr int HDIM  = 1024;
constexpr int VOCAB = 32000;
constexpr int BATCH = 64;
constexpr int SEQ   = 2048;
constexpr int G3H   = 3 * HDIM;     // 3072
constexpr int NCHUNK = 8;           // sequence chunks per batch row (flash split)
constexpr int CHUNK  = SEQ / NCHUNK;   // 256
constexpr int NTILE  = CHUNK / 8;      // 32 tiles of 8 rows per chunk

typedef __attribute__((ext_vector_type(2))) float v2f;
typedef __attribute__((ext_vector_type(4))) float v4f;
typedef __attribute__((ext_vector_type(8))) float v8f;

// ---------------------------------------------------------------------------
// WMMA helper: D = A(16x4 f32) * B(4x16 f32) + C(16x16 f32), wave32.
//   A: m = lane&15, VGPR j holds A[m][2*(lane>>4)+j]
//   B: n = lane&15, VGPR j holds B[2*(lane>>4)+j][n]
//   D: n = lane&15, VGPR r holds D[r + 8*(lane>>4)][n]
// ---------------------------------------------------------------------------
__device__ __forceinline__ v8f wmma4(v2f a, v2f b, v8f c) {
  return __builtin_amdgcn_wmma_f32_16x16x4_f32(
      false, a, false, b, (short)0, c, false, false);
}

__device__ __forceinline__ float wave_sum(float v) {
#pragma unroll
  for (int o = 16; o >= 1; o >>= 1) v += __shfl_xor(v, o, 32);
  return v;
}

// ---------------------------------------------------------------------------
// K1: gx = emb[seq] @ w_ih^T + b_ih ; gh = h @ w_hh^T + b_hh
// grid 96 x 128 (4 waves). 384 N-tiles (2 * 3072/16); wave = 1 N-tile x 4 M-tiles.
// ---------------------------------------------------------------------------
__global__ void k_gru_gemm(const int* __restrict__ seq,
                           const float* __restrict__ hprev,
                           const float* __restrict__ emb,
                           const float* __restrict__ w_ih,
                           const float* __restrict__ w_hh,
                           const float* __restrict__ b_ih,
                           const float* __restrict__ b_hh,
                           float* __restrict__ gx, float* __restrict__ gh) {
  const int wave = threadIdx.x >> 5;
  const int tile = blockIdx.x * 4 + wave;        // 0..383
  const int lane = threadIdx.x & 31;
  const int m = lane & 15, kh = lane >> 4, n = m;
  const bool isGh = tile >= (G3H / 16);
  const int n0 = (isGh ? tile - (G3H / 16) : tile) * 16;

  const float* w    = isGh ? w_hh : w_ih;
  const float* bias = isGh ? b_hh : b_ih;
  float* outp       = isGh ? gh : gx;

  const float* arow[4];
#pragma unroll
  for (int i = 0; i < 4; ++i) {
    arow[i] = isGh ? (hprev + (size_t)(i * 16 + m) * HDIM)
                   : (emb + (size_t)seq[i * 16 + m] * HDIM);
  }
  const float* brow = w + (size_t)(n0 + n) * HDIM;

  v8f acc[4] = {};
  for (int k0 = 0; k0 < HDIM; k0 += 4) {
    v2f bv = *(const v2f*)(brow + k0 + 2 * kh);
#pragma unroll
    for (int i = 0; i < 4; ++i) {
      v2f av = *(const v2f*)(arow[i] + k0 + 2 * kh);
      acc[i] = wmma4(av, bv, acc[i]);
    }
  }
  const float bs = bias[n0 + n];
#pragma unroll
  for (int i = 0; i < 4; ++i)
#pragma unroll
    for (int r = 0; r < 8; ++r)
      outp[(size_t)(i * 16 + r + 8 * kh) * G3H + n0 + n] = acc[i][r] + bs;
}

// ---------------------------------------------------------------------------
// K2: GRU gates -> h_new (PyTorch order r,z,n)
// ---------------------------------------------------------------------------
__global__ void k_gates(const float* __restrict__ gx, const float* __restrict__ gh,
                        const float* __restrict__ hprev, float* __restrict__ hnew) {
  const int idx = blockIdx.x * blockDim.x + threadIdx.x;
  const int b = idx >> 10, j = idx & (HDIM - 1);
  const size_t o = (size_t)b * G3H + j;
  const float r = 1.f / (1.f + expf(-(gx[o] + gh[o])));
  const float z = 1.f / (1.f + expf(-(gx[o + HDIM] + gh[o + HDIM])));
  const float nn = tanhf(gx[o + 2 * HDIM] + r * gh[o + 2 * HDIM]);
  const float hp = hprev[(size_t)b * HDIM + j];
  hnew[(size_t)b * HDIM + j] = (1.f - z) * nn + z * hp;
}

// ---------------------------------------------------------------------------
// K3: q = h_new @ attn_w  (64x1024x1024). grid 16 x 128.
// ---------------------------------------------------------------------------
__global__ void k_q_gemm(const float* __restrict__ hnew,
                         const float* __restrict__ attn_w,
                         float* __restrict__ q) {
  const int wave = threadIdx.x >> 5;
  const int tile = blockIdx.x * 4 + wave;        // 0..63
  const int lane = threadIdx.x & 31;
  const int m = lane & 15, kh = lane >> 4, n = m;
  const int n0 = tile * 16;

  const float* arow[4];
#pragma unroll
  for (int i = 0; i < 4; ++i) arow[i] = hnew + (size_t)(i * 16 + m) * HDIM;

  v8f acc[4] = {};
  for (int k0 = 0; k0 < HDIM; k0 += 4) {
    const int kk = k0 + 2 * kh;
    v2f bv = { attn_w[(size_t)kk * HDIM + n0 + n],
               attn_w[(size_t)(kk + 1) * HDIM + n0 + n] };
#pragma unroll
    for (int i = 0; i < 4; ++i) {
      v2f av = *(const v2f*)(arow[i] + k0 + 2 * kh);
      acc[i] = wmma4(av, bv, acc[i]);
    }
  }
#pragma unroll
  for (int i = 0; i < 4; ++i)
#pragma unroll
    for (int r = 0; r < 8; ++r)
      q[(size_t)(i * 16 + r + 8 * kh) * HDIM + n0 + n] = acc[i][r];
}

// K3b: hb[b] = h_new[b] . attn_b   (grid 64 x 32)
__global__ void k_hb(const float* __restrict__ hnew, const float* __restrict__ attn_b,
                     float* __restrict__ hb) {
  const int b = blockIdx.x, lane = threadIdx.x;
  float s = 0.f;
  for (int j = lane; j < HDIM; j += 32) s += hnew[(size_t)b * HDIM + j] * attn_b[j];
  s = wave_sum(s);
  if (lane == 0) hb[b] = s;
}

// ---------------------------------------------------------------------------
// K4: fused flash-style attention partial.
// grid (8 chunks, 64 batch) x 256 thr. Single pass over enc (536 MB total).
// Double-buffered 8-row x 4KB LDS tiles filled with global_load_async_to_lds_b128,
// pipelined with s_wait_asynccnt. Per chunk emits raw scores + (M, L, C) partial.
// ---------------------------------------------------------------------------
__global__ void __launch_bounds__(256)
k_attn_part(const float* __restrict__ q, const float* __restrict__ enc,
            const float* __restrict__ hb, float* __restrict__ scores_raw,
            float* __restrict__ Mpart, float* __restrict__ Lpart,
            float* __restrict__ Cpart) {
  const int chunk = blockIdx.x;          // 0..7
  const int b = blockIdx.y;              // 0..63
  const int tid = threadIdx.x;
  const int wid = tid >> 5, lane = tid & 31;
  const int s_base = chunk * CHUNK;

  __shared__ float buf[2][8 * HDIM];     // 64 KB double buffer (8 enc rows each)
  __shared__ float sc_tile[8];

  // q row cached in registers (same copy in every wave)
  v4f qreg[8];
  const float* qrow = q + (size_t)b * HDIM;
#pragma unroll
  for (int i = 0; i < 8; ++i) qreg[i] = *(const v4f*)(qrow + i * 128 + lane * 4);
  const float hbb = hb[b];

  // async-issue one 4KB enc row per wave into buf[t&1]
  auto issue = [&](int t) {
    const int s = s_base + t * 8 + wid;
    const float* g = enc + ((size_t)s * BATCH + b) * HDIM + lane * 4;
    const uint32_t lbase =
        (uint32_t)(uintptr_t)(&buf[t & 1][wid * HDIM + lane * 4]);
#pragma unroll
    for (int i = 0; i < 8; ++i) {
      asm volatile("global_load_async_to_lds_b128 %0, %1, off"
                   :: "v"(lbase + (uint32_t)(i * 512)),
                      "v"((uint64_t)(uintptr_t)(g + i * 128))
                   : "memory");
    }
  };

  issue(0);  // prologue

  float M = -3.0e38f, L = 0.f;
  v4f acc = {0.f, 0.f, 0.f, 0.f};

  for (int t = 0; t < NTILE; ++t) {
    if (t + 1 < NTILE) {
      issue(t + 1);
      asm volatile("s_wait_asynccnt 0x8" ::: "memory");  // tile t landed
    } else {
      asm volatile("s_wait_asynccnt 0x0" ::: "memory");
    }

    // score for this wave's own row (it issued it -> no barrier needed yet)
    const float* lrow = &buf[t & 1][wid * HDIM];
    float sum = 0.f;
#pragma unroll
    for (int i = 0; i < 8; ++i) {
      v4f e = *(const v4f*)(lrow + i * 128 + lane * 4);
      v4f qq = qreg[i];
      sum += e.x * qq.x + e.y * qq.y + e.z * qq.z + e.w * qq.w;
    }
    sum = wave_sum(sum);
    if (lane == 0) {
      sum += hbb;
      sc_tile[wid] = sum;
      scores_raw[(size_t)b * SEQ + (s_base + t * 8 + wid)] = sum;
    }
    __syncthreads();   // publish all 8 rows + scores to whole block

    // online softmax update (every thread computes identical M/L)
    float mt = sc_tile[0];
#pragma unroll
    for (int j = 1; j < 8; ++j) mt = fmaxf(mt, sc_tile[j]);
    const float newM = fmaxf(M, mt);
    const float f = expf(M - newM);
    float wj[8], sumw = 0.f;
#pragma unroll
    for (int j = 0; j < 8; ++j) { wj[j] = expf(sc_tile[j] - newM); sumw += wj[j]; }
    L = L * f + sumw;
    acc.x *= f; acc.y *= f; acc.z *= f; acc.w *= f;
#pragma unroll
    for (int j = 0; j < 8; ++j) {
      v4f e = *(const v4f*)(&buf[t & 1][j * HDIM + tid * 4]);
      acc.x += wj[j] * e.x; acc.y += wj[j] * e.y;
      acc.z += wj[j] * e.z; acc.w += wj[j] * e.w;
    }
    M = newM;
    __syncthreads();   // done reading buf[t&1] before it is re-filled
  }

  const int part = b * NCHUNK + chunk;
  if (tid == 0) { Mpart[part] = M; Lpart[part] = L; }
  *(v4f*)(Cpart + (size_t)part * HDIM + tid * 4) = acc;
}

// ---------------------------------------------------------------------------
// K5: combine chunk partials -> context + normalized attention weights.
// grid 64 x 256.
// ---------------------------------------------------------------------------
__global__ void k_attn_combine(const float* __restrict__ Mpart,
                               const float* __restrict__ Lpart,
                               const float* __restrict__ Cpart,
                               const float* __restrict__ scores_raw,
                               float* __restrict__ ctx, float* __restrict__ attn) {
  const int b = blockIdx.x, tid = threadIdx.x;
  float M = -3.0e38f;
#pragma unroll
  for (int c = 0; c < NCHUNK; ++c) M = fmaxf(M, Mpart[b * NCHUNK + c]);
  float L = 0.f, f[NCHUNK];
#pragma unroll
  for (int c = 0; c < NCHUNK; ++c) {
    f[c] = expf(Mpart[b * NCHUNK + c] - M);
    L += Lpart[b * NCHUNK + c] * f[c];
  }
  const float invL = 1.f / L;

  v4f acc = {0.f, 0.f, 0.f, 0.f};
#pragma unroll
  for (int c = 0; c < NCHUNK; ++c) {
    v4f e = *(const v4f*)(Cpart + (size_t)(b * NCHUNK + c) * HDIM + tid * 4);
    acc.x += f[c] * e.x; acc.y += f[c] * e.y;
    acc.z += f[c] * e.z; acc.w += f[c] * e.w;
  }
  acc.x *= invL; acc.y *= invL; acc.z *= invL; acc.w *= invL;
  *(v4f*)(ctx + (size_t)b * HDIM + tid * 4) = acc;

#pragma unroll
  for (int i = 0; i < 8; ++i) {
    const int s = tid + i * 256;
    attn[(size_t)b * SEQ + s] = expf(scores_raw[(size_t)b * SEQ + s] - M) * invL;
  }
}

// ---------------------------------------------------------------------------
// K6: co = tanh([h_new, ctx] @ concat_w^T + concat_b). grid 16 x 128.
// ---------------------------------------------------------------------------
__global__ void k_concat_gemm(const float* __restrict__ hnew, const float* __restrict__ ctx,
                              const float* __restrict__ concat_w,
                              const float* __restrict__ concat_b,
                              float* __restrict__ co) {
  const int wave = threadIdx.x >> 5;
  const int tile = blockIdx.x * 4 + wave;        // 0..63
  const int lane = threadIdx.x & 31;
  const int m = lane & 15, kh = lane >> 4, n = m;
  const int n0 = tile * 16;

  const float* ha[4]; const float* ca[4];
#pragma unroll
  for (int i = 0; i < 4; ++i) {
    ha[i] = hnew + (size_t)(i * 16 + m) * HDIM;
    ca[i] = ctx + (size_t)(i * 16 + m) * HDIM;
  }
  const float* brow = concat_w + (size_t)(n0 + n) * (2 * HDIM);

  v8f acc[4] = {};
  for (int k0 = 0; k0 < 2 * HDIM; k0 += 4) {
    v2f bv = *(const v2f*)(brow + k0 + 2 * kh);
    const bool lo = k0 < HDIM;
#pragma unroll
    for (int i = 0; i < 4; ++i) {
      const float* base = lo ? (ha[i] + k0) : (ca[i] + (k0 - HDIM));
      v2f av = *(const v2f*)(base + 2 * kh);
      acc[i] = wmma4(av, bv, acc[i]);
    }
  }
  const float bs = concat_b[n0 + n];
#pragma unroll
  for (int i = 0; i < 4; ++i)
#pragma unroll
    for (int r = 0; r < 8; ++r)
      co[(size_t)(i * 16 + r + 8 * kh) * HDIM + n0 + n] = tanhf(acc[i][r] + bs);
}

// ---------------------------------------------------------------------------
// K7: output = co @ out_w^T + out_b. 64x1024x32000. grid 500 x 128.
// ---------------------------------------------------------------------------
__global__ void k_out_gemm(const float* __restrict__ co, const float* __restrict__ out_w,
                           const float* __restrict__ out_b, float* __restrict__ out) {
  const int wave = threadIdx.x >> 5;
  const int tile = blockIdx.x * 4 + wave;        // 0..1999
  const int lane = threadIdx.x & 31;
  const int m = lane & 15, kh = lane >> 4, n = m;
  const int n0 = tile * 16;

  const float* arow[4];
#pragma unroll
  for (int i = 0; i < 4; ++i) arow[i] = co + (size_t)(i * 16 + m) * HDIM;
  const float* brow = out_w + (size_t)(n0 + n) * HDIM;

  v8f acc[4] = {};
  for (int k0 = 0; k0 < HDIM; k0 += 4) {
    v2f bv = *(const v2f*)(brow + k0 + 2 * kh);
#pragma unroll
    for (int i = 0; i < 4; ++i) {
      v2f av = *(const v2f*)(arow[i] + k0 + 2 * kh);
      acc[i] = wmma4(av, bv, acc[i]);
    }
  }
  const float bs = out_b[n0 + n];
#pragma unroll
  for (int i = 0; i < 4; ++i)
#pragma unroll
    for (int r = 0; r < 8; ++r)
      out[(size_t)(i * 16 + r + 8 * kh) * VOCAB + n0 + n] = acc[i][r] + bs;
}

// ---------------------------------------------------------------------------
extern "C" void kernel_launch(void* const* d_in, const int* in_sizes, int n_in,
                              void* d_out, int out_size, void* d_ws, size_t ws_size,
                              hipStream_t stream) {
  const int*   seq      = (const int*)d_in[0];
  const float* hprev    = (const float*)d_in[1];   // [1,64,1024]
  const float* enc      = (const float*)d_in[2];   // [2048,64,1024]
  const float* emb      = (const float*)d_in[3];   // [32000,1024]
  const float* w_ih     = (const float*)d_in[4];
  const float* w_hh     = (const float*)d_in[5];
  const float* b_ih     = (const float*)d_in[6];
  const float* b_hh     = (const float*)d_in[7];
  const float* attn_w   = (const float*)d_in[8];
  const float* attn_b   = (const float*)d_in[9];
  const float* concat_w = (const float*)d_in[10];
  const float* concat_b = (const float*)d_in[11];
  const float* out_w    = (const float*)d_in[12];
  const float* out_b    = (const float*)d_in[13];

  float* out  = (float*)d_out;                               // [64,32000]
  float* hnew = out + (size_t)BATCH * VOCAB;                 // [64,1024]
  float* attn = hnew + (size_t)BATCH * HDIM;                 // [64,2048]

  float* w = (float*)d_ws;
  float* gx     = w;  w += (size_t)BATCH * G3H;
  float* gh     = w;  w += (size_t)BATCH * G3H;
  float* q      = w;  w += (size_t)BATCH * HDIM;
  float* hb     = w;  w += BATCH;
  float* sraw   = w;  w += (size_t)BATCH * SEQ;
  float* Mpart  = w;  w += (size_t)BATCH * NCHUNK;
  float* Lpart  = w;  w += (size_t)BATCH * NCHUNK;
  float* Cpart  = w;  w += (size_t)BATCH * NCHUNK * HDIM;
  float* ctx    = w;  w += (size_t)BATCH * HDIM;
  float* co     = w;  w += (size_t)BATCH * HDIM;

  hipLaunchKernelGGL(k_gru_gemm, dim3(96), dim3(128), 0, stream,
                     seq, hprev, emb, w_ih, w_hh, b_ih, b_hh, gx, gh);
  hipLaunchKernelGGL(k_gates, dim3(256), dim3(256), 0, stream, gx, gh, hprev, hnew);
  hipLaunchKernelGGL(k_q_gemm, dim3(16), dim3(128), 0, stream, hnew, attn_w, q);
  hipLaunchKernelGGL(k_hb, dim3(64), dim3(32), 0, stream, hnew, attn_b, hb);
  hipLaunchKernelGGL(k_attn_part, dim3(NCHUNK, BATCH), dim3(256), 0, stream,
                     q, enc, hb, sraw, Mpart, Lpart, Cpart);
  hipLaunchKernelGGL(k_attn_combine, dim3(BATCH), dim3(256), 0, stream,
                     Mpart, Lpart, Cpart, sraw, ctx, attn);
  hipLaunchKernelGGL(k_concat_gemm, dim3(16), dim3(128), 0, stream,
                     hnew, ctx, concat_w, concat_b, co);
  hipLaunchKernelGGL(k_out_gemm, dim3(500), dim3(128), 0, stream, co, out_w, out_b, out);
}